// SimpleBiLSTM_1657857376430
// MI455X (gfx1250) — compile-verified
//
#include <hip/hip_runtime.h>
#include <hip/hip_bf16.h>

// BiLSTM: B=32, T=512, D=512, H=512, fp32, flax gate order i,f,g,o.
// Persistent fused kernel: per step, y = [x_t | h] @ [Wi; Wh] + b via
// V_WMMA_F32_16X16X4_F32 with weights repacked into WMMA fragment order
// (two coalesced b128 loads per K-step). Branch-free activations, b128
// staging, register-resident cell state, device-wide step barrier.

#define Bsz 32
#define Tlen 512
#define Din 512
#define Hh 512
#define G4 2048          // 4*H
#define NBLK 16          // 8 blocks per direction
#define LDS_STRIDE 1028  // 1024 + 4 pad -> conflict-free fragment reads

// packed-W geometry: per (dir, jt) slab of 256KB:
//   [kb4 (256)][kk (2)][n (16)][gate (4)][p (2)]  floats
#define WP_SLAB (256 * 2 * 16 * 4 * 2)     // 65536 floats per (dir,jt)
#define WP_TOTAL (2 * 32 * WP_SLAB)        // 4,194,304 floats = 16 MB

typedef __attribute__((ext_vector_type(2))) float v2f;
typedef __attribute__((ext_vector_type(4))) float v4f;
typedef __attribute__((ext_vector_type(8))) float v8f;

__global__ void init_ws_kernel(float* h, unsigned* cnt) {
    int i = blockIdx.x * blockDim.x + threadIdx.x;
    if (i < 2 * Bsz * Hh) h[i] = 0.0f;
    if (i == 0) *cnt = 0u;
}

// Repack [Wi;Wh] (K=1024 x 2048) for both directions into WMMA fragment order.
__global__ void repack_w_kernel(const float* __restrict__ Wi_f,
                                const float* __restrict__ Wh_f,
                                const float* __restrict__ Wi_b,
                                const float* __restrict__ Wh_b,
                                float* __restrict__ Wp) {
    int idx = blockIdx.x * blockDim.x + threadIdx.x;  // < WP_TOTAL
    int p   = idx & 1;
    int g   = (idx >> 1) & 3;
    int n   = (idx >> 3) & 15;
    int kk  = (idx >> 7) & 1;
    int kb4 = (idx >> 8) & 255;
    int jt  = (idx >> 16) & 31;
    int d   = idx >> 21;
    int row = kb4 * 4 + 2 * kk + p;        // 0..1023 over [Wi;Wh]
    int col = g * Hh + jt * 16 + n;        // 0..2047
    const float* Wi = d ? Wi_b : Wi_f;
    const float* Wh = d ? Wh_b : Wh_f;
    float v = (row < Din) ? Wi[row * G4 + col] : Wh[(row - Din) * G4 + col];
    Wp[idx] = v;
}

// branch-free: sigmoid(x) = rcp(1 + exp(-x))
__device__ __forceinline__ float sigmoid_(float x) {
    return __builtin_amdgcn_rcpf(1.0f + __expf(-x));
}
// branch-free: tanh(x) = sign(x) * (1 - 2*rcp(exp(2|x|) + 1))
__device__ __forceinline__ float tanh_(float x) {
    float e = __expf(2.0f * __builtin_fabsf(x));
    float t = 1.0f - 2.0f * __builtin_amdgcn_rcpf(e + 1.0f);
    return __builtin_copysignf(t, x);
}

__global__ __launch_bounds__(256, 1)
void bilstm_persistent(const float* __restrict__ X,
                       const long long* __restrict__ lengths,
                       const float* __restrict__ b_f,
                       const float* __restrict__ b_b,
                       const float* __restrict__ Wp,
                       float* __restrict__ out,      // [B, T, 2H]
                       float* __restrict__ h_buf,    // [2, B, H]
                       unsigned* __restrict__ cnt) {
    extern __shared__ float lds[];      // 16 rows x LDS_STRIDE floats
    __shared__ int s_ts[16];            // per-row source timestep (gather)

    const int tid  = threadIdx.x;
    const int lane = tid & 31;
    const int wave = tid >> 5;
    const int blk  = blockIdx.x;
    const int dir  = blk >> 3;            // 0 = fwd, 1 = bwd
    const int u    = blk & 7;
    const int unit = u * 8 + wave;        // 0..63 per direction
    const int mt   = unit >> 5;           // 0..1 (batch tile)
    const int jt   = unit & 31;           // 0..31 (hidden-column tile)
    const int mbase = mt * 16;            // all waves in a block share mt

    const float* __restrict__ bias = dir ? b_b : b_f;
    float* __restrict__ h = h_buf + dir * (Bsz * Hh);

    const int n  = lane & 15;             // N within tile (and A-row M)
    const int kk = lane >> 4;             // K-pair select (0: K0/1, 1: K2/3)
    const int j  = jt * 16 + n;           // hidden column 0..511

    const float bia = bias[0 * Hh + j];
    const float bif = bias[1 * Hh + j];
    const float big = bias[2 * Hh + j];
    const float bio = bias[3 * Hh + j];

    // per-lane output flip base (len-1) for the 8 rows this lane owns
    int l1[8];
#pragma unroll
    for (int r = 0; r < 8; ++r) l1[r] = (int)lengths[mbase + r + 8 * kk] - 1;

    float c[8];
#pragma unroll
    for (int r = 0; r < 8; ++r) c[r] = 0.0f;

    const float* lA = lds + (lane & 15) * LDS_STRIDE;
    // this wave's lane-base into its (dir, jt) weight slab
    const float* wp = Wp + ((size_t)(dir * 32 + jt) * WP_SLAB) + ((kk * 16 + n) << 3);

    // source timestep for step 0
    if (tid < 16) {
        int ts = 0;
        if (dir != 0) {
            int q = (int)lengths[mbase + tid] - 1;
            if (q < 0) q += Tlen;
            ts = q;
        }
        s_ts[tid] = ts;
    }
    __syncthreads();

    for (int t = 0; t < Tlen; ++t) {
        // ---- stage A = [x_t | h_{t-1}] for 16 batch rows, b128 granular ----
        for (int q = tid; q < 16 * 128; q += 256) {       // x half: 32KB
            const int row = q >> 7;
            const int col = (q & 127) << 2;
            const v4f v = *(const v4f*)(X + ((long)(mbase + row) * Tlen + s_ts[row]) * Din + col);
            *(v4f*)(lds + row * LDS_STRIDE + col) = v;
        }
        for (int q = tid; q < 16 * 128; q += 256) {       // h half: 32KB
            const int row = q >> 7;
            const int col = (q & 127) << 2;
            const v4f v = *(const v4f*)(h + (mbase + row) * Hh + col);
            *(v4f*)(lds + row * LDS_STRIDE + Din + col) = v;
        }
        __syncthreads();

        // ---- prefetch next timestep's x rows (256 x 128B lines = 32KB) ----
        {
            const int tn  = (t + 1 < Tlen) ? t + 1 : t;
            const int row = tid >> 4;            // 16 lines per 2KB row
            const int off = (tid & 15) * 32;     // 128B line in floats
            const int b   = mbase + row;
            int ts = tn;
            if (dir != 0) {
                int q = (int)lengths[b] - 1 - tn;
                if (q < 0) q += Tlen;
                ts = q;
            }
            __builtin_prefetch(&X[((long)b * Tlen + ts) * Din + off], 0, 1);
        }

        v8f acci = {}, accf = {}, accg = {}, acco = {};

        // ---- fused K-loop over [Wi; Wh], K = 1024, packed B fragments ----
#pragma unroll 4
        for (int kb4 = 0; kb4 < 256; ++kb4) {
            const v2f a   = *(const v2f*)(lA + kb4 * 4 + 2 * kk);
            const v4f w01 = *(const v4f*)(wp + kb4 * 256);      // gates i,f
            const v4f w23 = *(const v4f*)(wp + kb4 * 256 + 4);  // gates g,o
            v2f bvi = {w01[0], w01[1]};
            v2f bvf = {w01[2], w01[3]};
            v2f bvg = {w23[0], w23[1]};
            v2f bvo = {w23[2], w23[3]};
            acci = __builtin_amdgcn_wmma_f32_16x16x4_f32(false, a, false, bvi, (short)0, acci, false, false);
            accf = __builtin_amdgcn_wmma_f32_16x16x4_f32(false, a, false, bvf, (short)0, accf, false, false);
            accg = __builtin_amdgcn_wmma_f32_16x16x4_f32(false, a, false, bvg, (short)0, accg, false, false);
            acco = __builtin_amdgcn_wmma_f32_16x16x4_f32(false, a, false, bvo, (short)0, acco, false, false);
        }

        // ---- cell update for the 8 (row, j) elements this lane owns ----
#pragma unroll
        for (int r = 0; r < 8; ++r) {
            const int b = mbase + r + 8 * kk;   // C/D: VGPR r -> M=r (+8 upper half)
            const float ig = sigmoid_(acci[r] + bia);
            const float fg = sigmoid_(accf[r] + bif);
            const float gg = tanh_(accg[r] + big);
            const float og = sigmoid_(acco[r] + bio);
            const float c2 = fg * c[r] + ig * gg;
            c[r] = c2;
            const float h2 = og * tanh_(c2);
            h[b * Hh + j] = h2;
            int tout = t;
            if (dir != 0) {
                int q = l1[r] - t;              // output flip = same involution
                if (q < 0) q += Tlen;
                tout = q;
            }
            out[((long)b * Tlen + tout) * (2 * Hh) + dir * Hh + j] = h2;
        }

        // ---- publish next step's gather indices (separate shared array) ----
        __syncthreads();
        if (tid < 16) {
            const int tn = t + 1;
            int ts = tn;
            if (dir != 0) {
                int q = (int)lengths[mbase + tid] - 1 - tn;
                if (q < 0) q += Tlen;
                ts = q;
            }
            s_ts[tid] = (tn < Tlen) ? ts : 0;
        }

        // ---- device-wide step barrier (monotonic counter, no reset) ----
        if (tid == 0) {
            __threadfence();                    // release h writes
            atomicAdd(cnt, 1u);
            const unsigned target = (unsigned)(t + 1) * NBLK;
            while (__atomic_load_n(cnt, __ATOMIC_RELAXED) < target) {
                __builtin_amdgcn_s_sleep(2);
            }
        }
        __syncthreads();
        __threadfence();                        // acquire other blocks' h
    }
}

extern "C" void kernel_launch(void* const* d_in, const int* in_sizes, int n_in,
                              void* d_out, int out_size, void* d_ws, size_t ws_size,
                              hipStream_t stream) {
    const float*     X       = (const float*)d_in[0];
    const long long* lengths = (const long long*)d_in[1];
    const float*     Wi_f    = (const float*)d_in[2];
    const float*     Wh_f    = (const float*)d_in[3];
    const float*     b_f     = (const float*)d_in[4];
    const float*     Wi_b    = (const float*)d_in[5];
    const float*     Wh_b    = (const float*)d_in[6];
    const float*     b_b     = (const float*)d_in[7];
    float*           out     = (float*)d_out;

    // workspace layout: h (128KB) | cnt | pad | Wpack (16MB)
    float*    h_buf = (float*)d_ws;
    unsigned* cnt   = (unsigned*)((char*)d_ws + 2 * Bsz * Hh * sizeof(float));
    float*    Wp    = (float*)((char*)d_ws + 2 * Bsz * Hh * sizeof(float) + 4096);

    const int init_n = 2 * Bsz * Hh;
    init_ws_kernel<<<(init_n + 255) / 256, 256, 0, stream>>>(h_buf, cnt);

    repack_w_kernel<<<WP_TOTAL / 256, 256, 0, stream>>>(Wi_f, Wh_f, Wi_b, Wh_b, Wp);

    const size_t lds_bytes = (size_t)16 * LDS_STRIDE * sizeof(float);
    bilstm_persistent<<<NBLK, 256, lds_bytes, stream>>>(
        X, lengths, b_f, b_b, Wp, out, h_buf, cnt);
}